// SimpleOnsetEdgePooling_82308753260715
// MI455X (gfx1250) — compile-verified
//
#include <hip/hip_runtime.h>
#include <math.h>

// CDNA5 / gfx1250: wave32. WMMA 16x16x4 F32 fragment types:
//   A (16x4 f32)  = 64 vals / 32 lanes = 2 VGPRs -> v2f
//   C/D (16x16)   = 256 vals / 32 lanes = 8 VGPRs -> v8f
typedef __attribute__((ext_vector_type(2))) float v2f;
typedef __attribute__((ext_vector_type(8))) float v8f;

#define C_DIM 128
#define CAT   256   // 2*C (concat of src,dst features)

// ---------------------------------------------------------------------------
// Kernel A: edge scores for the m merged edges via V_WMMA_F32_16X16X4_F32.
// One wave per tile of 16 edges. Stage concat(x[s],x[t]) rows into LDS with
// coalesced float4 (global_load_b128) gathers, then 64 K=4 WMMA steps with the
// weight chunk in column 0 of B; D column 0 = the 16 dot products.
//
// Branchless fragment construction: both A and B fragments are single aligned
// float2 LDS loads (low lanes take K0,K1; high lanes K2,K3); B lanes with
// N != 0 are zeroed with a value select (v_cndmask), never a predicated load,
// so EXEC stays all-1s through the whole WMMA loop with no saveexec churn.
// ---------------------------------------------------------------------------
__global__ __launch_bounds__(32)
void edge_score_wmma(const float* __restrict__ x,
                     const float* __restrict__ w,      // 256 weights
                     const float* __restrict__ bias,   // 1 bias
                     const int* __restrict__ src,
                     const int* __restrict__ dst,
                     const int* __restrict__ merged,
                     int m,
                     float* __restrict__ score) {
  __shared__ float tile[16 * CAT];   // 16 KB: [row 0..15][channel 0..255]
  __shared__ float wl[CAT];          // staged weights
  __shared__ int   s_row[16], t_row[16];

  const int lane  = threadIdx.x;
  const int tile0 = blockIdx.x * 16;

  if (lane < 16) {
    int k = tile0 + lane;
    int e = merged[(k < m) ? k : 0];   // OOB rows score garbage, never stored
    s_row[lane] = src[e];
    t_row[lane] = dst[e];
  }
  // stage weights (64 float4 across 32 lanes)
  {
    const float4* w4 = (const float4*)w;
    for (int j = lane; j < CAT / 4; j += 32) ((float4*)wl)[j] = w4[j];
  }
  __syncthreads();

  // Stage 16 gathered rows: row r = concat(x[s_r], x[t_r]) (256 f32 each).
  // j sweeps 16 rows x 64 float4; consecutive lanes hit consecutive 16B
  // chunks of one source row -> fully coalesced 512B bursts.
  const float4* x4 = (const float4*)x;
  for (int j = lane; j < 16 * 64; j += 32) {
    int r  = j >> 6;       // row
    int cq = j & 63;       // float4 index within concat row
    float4 v = (cq < 32) ? x4[(size_t)s_row[r] * 32 + cq]
                         : x4[(size_t)t_row[r] * 32 + (cq - 32)];
    ((float4*)tile)[r * 64 + cq] = v;
  }
  __syncthreads();

  const bool hi    = (lane >= 16);
  const int  rlow  = lane & 15;
  const bool isN0  = (rlow == 0);      // lanes holding column N==0 of B
  const int  koff  = hi ? 2 : 0;       // K-pair selector for this half-wave

  // Per-lane base pointers; loop only bumps them by 4 floats.
  const float* ap = &tile[rlow * CAT + koff];
  const float* wp = &wl[koff];

  v8f acc = {0.f, 0.f, 0.f, 0.f, 0.f, 0.f, 0.f, 0.f};

  // ISA A-layout (16x4 f32): lanes 0-15 hold row M=lane, (K0,K1) in (v0,v1);
  // lanes 16-31 hold (K2,K3). B-layout (4x16): VGPR0 lanes0-15=row K0,
  // lanes16-31=row K2; VGPR1=K1/K3; only column N==0 (lanes 0 and 16) nonzero.
#pragma unroll 4
  for (int k0 = 0; k0 < CAT; k0 += 4) {
    v2f a; a.x = ap[k0]; a.y = ap[k0 + 1];     // one aligned ds_load_b64

    float w0 = wp[k0];                          // one aligned ds_load_b64
    float w1 = wp[k0 + 1];
    v2f b;
    b.x = isN0 ? w0 : 0.0f;                     // v_cndmask, no EXEC change
    b.y = isN0 ? w1 : 0.0f;

    acc = __builtin_amdgcn_wmma_f32_16x16x4_f32(
        /*neg_a=*/false, a, /*neg_b=*/false, b,
        /*c_mod=*/(short)0, acc, /*reuse_a=*/false, /*reuse_b=*/false);
  }

  // D column N=0: lane 0 holds M=0..7 in acc[0..7], lane 16 holds M=8..15.
  if (isN0) {
    float bv   = bias[0];
    int   base = tile0 + (hi ? 8 : 0);
#pragma unroll
    for (int v = 0; v < 8; ++v) {
      int k = base + v;
      if (k < m) {
        float e = acc[v] + bv;
        score[k] = 1.0f / (1.0f + __expf(-e)) + 0.5f;
      }
    }
  }
}

// ---------------------------------------------------------------------------
// Kernel B: merged-cluster rows. out[k] = (x[s] + (s!=t ? x[t] : 0)) * score[k]
// One wave per merged edge; lane = one float4 chunk (coalesced b128 traffic).
// ---------------------------------------------------------------------------
__global__ __launch_bounds__(32)
void merged_rows(const float* __restrict__ x,
                 const int* __restrict__ src, const int* __restrict__ dst,
                 const int* __restrict__ merged,
                 const float* __restrict__ score,
                 float* __restrict__ out, int m) {
  int k = blockIdx.x;
  if (k >= m) return;
  int e = merged[k];
  int s = src[e], t = dst[e];
  int lane = threadIdx.x;

  const float4* x4 = (const float4*)x;
  float4 a = x4[(size_t)s * 32 + lane];
  if (s != t) {
    float4 b = x4[(size_t)t * 32 + lane];
    a.x += b.x; a.y += b.y; a.z += b.z; a.w += b.w;
  }
  float sc = score[k];
  float4 o; o.x = a.x * sc; o.y = a.y * sc; o.z = a.z * sc; o.w = a.w * sc;
  ((float4*)out)[(size_t)k * 32 + lane] = o;
}

// ---------------------------------------------------------------------------
// Kernel C: singleton clusters (score == 1): out[cluster[i]] = x[i] for
// cluster[i] >= m. Exactly one node maps to each such row -> no atomics.
// 8 waves / block, one node per wave, float4 per lane.
// ---------------------------------------------------------------------------
__global__ __launch_bounds__(256)
void singleton_rows(const float* __restrict__ x,
                    const int* __restrict__ cluster,
                    float* __restrict__ out, int n, int m) {
  int wave = threadIdx.x >> 5;
  int lane = threadIdx.x & 31;
  int i = blockIdx.x * 8 + wave;
  if (i >= n) return;
  int k = cluster[i];
  if (k < m) return;   // handled by merged_rows
  const float4* x4 = (const float4*)x;
  ((float4*)out)[(size_t)k * 32 + lane] = x4[(size_t)i * 32 + lane];
}

// ---------------------------------------------------------------------------
// Launch. Inputs (reference order): x[f32 N*128], lin_w[f32 256], lin_b[f32 1],
// edge_index[i32 2*E], cluster[i32 N], merged_edges[i32 m], num_clusters[i32 1]
// Output: [K,128] f32, K = out_size/128. Scratch: score[m] f32 in d_ws.
// Every output row is written exactly once (k<m by B, k>=m by C), so no
// zero-init of d_out is needed.
// ---------------------------------------------------------------------------
extern "C" void kernel_launch(void* const* d_in, const int* in_sizes, int n_in,
                              void* d_out, int out_size, void* d_ws, size_t ws_size,
                              hipStream_t stream) {
  const float* x       = (const float*)d_in[0];
  const float* lin_w   = (const float*)d_in[1];
  const float* lin_b   = (const float*)d_in[2];
  const int*   eidx    = (const int*)d_in[3];
  const int*   cluster = (const int*)d_in[4];
  const int*   merged  = (const int*)d_in[5];

  const int n = in_sizes[0] / C_DIM;   // 100000
  const int E = in_sizes[3] / 2;       // 500000
  const int m = in_sizes[5];           // merged edge count

  const int* src = eidx;
  const int* dst = eidx + E;

  float* out   = (float*)d_out;
  float* score = (float*)d_ws;         // m floats of scratch

  if (m > 0) {
    int tiles = (m + 15) / 16;
    edge_score_wmma<<<tiles, 32, 0, stream>>>(x, lin_w, lin_b, src, dst,
                                              merged, m, score);
    merged_rows<<<m, 32, 0, stream>>>(x, src, dst, merged, score, out, m);
  }
  if (n > 0) {
    singleton_rows<<<(n + 7) / 8, 256, 0, stream>>>(x, cluster, out, n, m);
  }
}